// GATv2Layer_20890720928259
// MI455X (gfx1250) — compile-verified
//
#include <hip/hip_runtime.h>
#include <hip/hip_bf16.h>
#include <math.h>

// ---------------- problem constants (from reference) ----------------
#define NN 50000
#define EE 800000
#define FF 128        // == H*D
#define HH 8
#define DD 16

typedef __attribute__((ext_vector_type(2))) float v2f;
typedef __attribute__((ext_vector_type(8))) float v8f;

// ---------------------------------------------------------------------------
// K0: zero a float buffer
// ---------------------------------------------------------------------------
__global__ void gat_zero_kernel(float* __restrict__ p, int n) {
    int i = blockIdx.x * blockDim.x + threadIdx.x;
    if (i < n) p[i] = 0.0f;
}

// ---------------------------------------------------------------------------
// K1: projection GEMM via V_WMMA_F32_16X16X4_F32 (full fp32 matrix pipe).
// One wave (32 lanes) computes a 16-row x 128-col panel of (x @ W) for
// either W_src (blockIdx.x==0) or W_dst (blockIdx.x==1).
//
// ISA 7.12.2 fp32 layouts (wave32):
//   A 16x4 : lanes 0-15 row M=l, VGPR0=K0, VGPR1=K1; lanes 16-31 K2/K3
//   B 4x16 : VGPR0 = row K0 (lanes 0-15) / K2 (lanes 16-31), VGPR1 = K1/K3,
//            column N = lane&15
//   C 16x16: VGPR r = row r (lanes 0-15) / row r+8 (lanes 16-31), col = lane&15
// ---------------------------------------------------------------------------
__global__ __launch_bounds__(32) void gat_proj_gemm(
    const float* __restrict__ x,
    const float* __restrict__ W_src,
    const float* __restrict__ W_dst,
    float* __restrict__ src_p,
    float* __restrict__ dst_p)
{
    const int which   = blockIdx.x;            // 0 -> W_src, 1 -> W_dst
    const int rowbase = blockIdx.y * 16;       // NN % 16 == 0
    const float* __restrict__ W   = which ? W_dst : W_src;
    float* __restrict__       out = which ? dst_p : src_p;

    const int lane = threadIdx.x;
    const int half = lane >> 4;                // 0: K0/K1 side, 1: K2/K3 side
    const int l    = lane & 15;

    const float* __restrict__ xrow = x + (size_t)(rowbase + l) * FF;

    v8f acc[8];
#pragma unroll
    for (int c = 0; c < 8; ++c) acc[c] = (v8f){};

    for (int k = 0; k < FF; k += 4) {
        v2f a;
        a.x = xrow[k + 2 * half + 0];
        a.y = xrow[k + 2 * half + 1];
#pragma unroll
        for (int c = 0; c < 8; ++c) {
            const float* wp = W + (size_t)(k + 2 * half) * FF + c * 16 + l;
            v2f b;
            b.x = wp[0];       // K = k + 2*half
            b.y = wp[FF];      // K = k + 2*half + 1
            // 8 args: (neg_a, A, neg_b, B, c_mod, C, reuse_a, reuse_b)
            acc[c] = __builtin_amdgcn_wmma_f32_16x16x4_f32(
                false, a, false, b, (short)0, acc[c], false, false);
        }
    }

#pragma unroll
    for (int c = 0; c < 8; ++c) {
#pragma unroll
        for (int r = 0; r < 8; ++r) {
            out[(size_t)(rowbase + r + 8 * half) * FF + c * 16 + l] = acc[c][r];
        }
    }
}

// ---------------------------------------------------------------------------
// K2: per-(edge,head) GATv2 score + per-block max (for global softmax max)
// ---------------------------------------------------------------------------
__global__ __launch_bounds__(256) void gat_score(
    const float* __restrict__ src_p,
    const float* __restrict__ dst_p,
    const int*   __restrict__ srcI,
    const int*   __restrict__ dstI,
    const float* __restrict__ attn,
    float* __restrict__ scorebuf,
    float* __restrict__ partial)
{
    __shared__ float smax[256];
    const int tid = blockIdx.x * 256 + threadIdx.x;
    float s = -INFINITY;
    if (tid < EE * HH) {
        const int e = tid >> 3;
        const int h = tid & 7;
        const float* sp = src_p + (size_t)srcI[e] * FF + h * DD;
        const float* dp = dst_p + (size_t)dstI[e] * FF + h * DD;
        const float* at = attn + h * DD;
        float acc = 0.0f;
#pragma unroll
        for (int d = 0; d < DD; ++d) {
            float v = sp[d] + dp[d];
            v = (v >= 0.0f) ? v : 0.2f * v;    // leaky_relu BEFORE attn dot (GATv2)
            acc += at[d] * v;
        }
        scorebuf[tid] = acc;
        s = acc;
    }
    smax[threadIdx.x] = s;
    __syncthreads();
#pragma unroll
    for (int off = 128; off > 0; off >>= 1) {
        if (threadIdx.x < off)
            smax[threadIdx.x] = fmaxf(smax[threadIdx.x], smax[threadIdx.x + off]);
        __syncthreads();
    }
    if (threadIdx.x == 0) partial[blockIdx.x] = smax[0];
}

// ---------------------------------------------------------------------------
// K3: reduce block maxima to the single global max
// ---------------------------------------------------------------------------
__global__ __launch_bounds__(256) void gat_maxreduce(
    const float* __restrict__ partial, int n, float* __restrict__ gmax)
{
    __shared__ float smax[256];
    float s = -INFINITY;
    for (int i = threadIdx.x; i < n; i += 256) s = fmaxf(s, partial[i]);
    smax[threadIdx.x] = s;
    __syncthreads();
#pragma unroll
    for (int off = 128; off > 0; off >>= 1) {
        if (threadIdx.x < off)
            smax[threadIdx.x] = fmaxf(smax[threadIdx.x], smax[threadIdx.x + off]);
        __syncthreads();
    }
    if (threadIdx.x == 0) *gmax = smax[0];
}

// ---------------------------------------------------------------------------
// K4: exp(score - gmax) in place + denom[dst,h] += exp  (atomic, L2-resident)
// ---------------------------------------------------------------------------
__global__ __launch_bounds__(256) void gat_expdenom(
    float* __restrict__ scorebuf,
    const int* __restrict__ dstI,
    const float* __restrict__ gmax,
    float* __restrict__ denom)
{
    const int tid = blockIdx.x * 256 + threadIdx.x;
    if (tid >= EE * HH) return;
    const int e = tid >> 3;
    const int h = tid & 7;
    const float ex = __expf(scorebuf[tid] - *gmax);
    scorebuf[tid] = ex;
    atomicAdd(&denom[(size_t)dstI[e] * HH + h], ex);
}

// ---------------------------------------------------------------------------
// K5: message passing: out[dst,h,:] += src_p[src,h,:] * alpha
// ---------------------------------------------------------------------------
__global__ __launch_bounds__(256) void gat_message(
    const float* __restrict__ src_p,
    const float* __restrict__ expbuf,
    const float* __restrict__ denom,
    const int*   __restrict__ srcI,
    const int*   __restrict__ dstI,
    float* __restrict__ out)
{
    const int tid = blockIdx.x * 256 + threadIdx.x;
    if (tid >= EE * HH) return;
    const int e = tid >> 3;
    const int h = tid & 7;
    const int di = dstI[e];
    const float alpha = expbuf[tid] / (denom[(size_t)di * HH + h] + 1e-16f);
    const float* sp = src_p + (size_t)srcI[e] * FF + h * DD;
    float* op = out + (size_t)di * FF + h * DD;
#pragma unroll
    for (int d = 0; d < DD; ++d) {
        atomicAdd(&op[d], sp[d] * alpha);   // global_atomic_add_f32 (no return)
    }
}

// ---------------------------------------------------------------------------
// K6: residual + bias + PReLU, in place on d_out
// ---------------------------------------------------------------------------
__global__ __launch_bounds__(256) void gat_finalize(
    float* __restrict__ out,
    const float* __restrict__ x,
    const float* __restrict__ bias,
    const float* __restrict__ prelu_w)
{
    const int i = blockIdx.x * 256 + threadIdx.x;
    if (i >= NN * FF) return;
    const float v  = out[i] + x[i] + bias[i & (FF - 1)];
    const float pw = prelu_w[0];
    out[i] = (v >= 0.0f) ? v : pw * v;
}

// ---------------------------------------------------------------------------
// launch
// ---------------------------------------------------------------------------
extern "C" void kernel_launch(void* const* d_in, const int* in_sizes, int n_in,
                              void* d_out, int out_size, void* d_ws, size_t ws_size,
                              hipStream_t stream) {
    (void)in_sizes; (void)n_in; (void)out_size; (void)ws_size;

    const float* x        = (const float*)d_in[0];
    const int*   eidx     = (const int*)  d_in[1];
    const float* W_src    = (const float*)d_in[2];
    const float* W_dst    = (const float*)d_in[3];
    const float* attn     = (const float*)d_in[4];
    const float* bias     = (const float*)d_in[5];
    const float* prelu_w  = (const float*)d_in[6];
    const int*   srcI     = eidx;          // edge_index[0,:]
    const int*   dstI     = eidx + EE;     // edge_index[1,:]

    float* out = (float*)d_out;            // [N, H*D] fp32, used as accumulator

    // workspace layout (floats)
    float* ws      = (float*)d_ws;
    float* src_p   = ws;                                   // N*F
    float* dst_p   = src_p + (size_t)NN * FF;              // N*F
    float* scoreb  = dst_p + (size_t)NN * FF;              // E*H (score -> exp)
    float* denom   = scoreb + (size_t)EE * HH;             // N*H
    float* partial = denom + (size_t)NN * HH;              // nScoreBlk
    const int nScoreBlk = (EE * HH + 255) / 256;           // 25000
    float* gmax    = partial + nScoreBlk;                  // 1

    // K0: zero accumulators
    {
        const int n1 = NN * FF;
        gat_zero_kernel<<<(n1 + 255) / 256, 256, 0, stream>>>(out, n1);
        const int n2 = NN * HH;
        gat_zero_kernel<<<(n2 + 255) / 256, 256, 0, stream>>>(denom, n2);
    }

    // K1: fp32 WMMA projections (src_p, dst_p)
    {
        dim3 grid(2, NN / 16);             // 50000 % 16 == 0
        gat_proj_gemm<<<grid, 32, 0, stream>>>(x, W_src, W_dst, src_p, dst_p);
    }

    // K2: scores + block max
    gat_score<<<nScoreBlk, 256, 0, stream>>>(src_p, dst_p, srcI, dstI, attn,
                                             scoreb, partial);
    // K3: global max
    gat_maxreduce<<<1, 256, 0, stream>>>(partial, nScoreBlk, gmax);

    // K4: exp + denom atomics
    gat_expdenom<<<nScoreBlk, 256, 0, stream>>>(scoreb, dstI, gmax, denom);

    // K5: weighted message scatter-add
    gat_message<<<nScoreBlk, 256, 0, stream>>>(src_p, scoreb, denom, srcI, dstI, out);

    // K6: residual + bias + PReLU
    {
        const int n = NN * FF;
        gat_finalize<<<(n + 255) / 256, 256, 0, stream>>>(out, x, bias, prelu_w);
    }
}